// HeteroSAGE_30365418783288
// MI455X (gfx1250) — compile-verified
//
#include <hip/hip_runtime.h>
#include <hip/hip_bf16.h>

// ---------------------------------------------------------------------------
// HeteroSAGE (2-layer, 4-relation) for MI455X / gfx1250, wave32.
//  - mean-aggregation via f32 atomic scatter (L2-resident destinations)
//  - layer-0 GEMMs: async global->LDS staged A panels, M-blocked (2 tiles)
//    + V_WMMA_F32_16X16X4_F32 with two independent accumulators
//  - layer-1 (dout=2) via bandwidth-bound float4 dot kernel
// ---------------------------------------------------------------------------

typedef __attribute__((ext_vector_type(2))) float v2f;
typedef __attribute__((ext_vector_type(8))) float v8f;
typedef int v4i __attribute__((vector_size(16)));

#define AS1 __attribute__((address_space(1)))
#define AS3 __attribute__((address_space(3)))

#if __has_builtin(__builtin_amdgcn_global_load_async_to_lds_b128)
#define HAVE_ASYNC_LDS 1
#else
#define HAVE_ASYNC_LDS 0
#endif

#define NN   50000
#define EE   400000
#define DIN  128
#define DH   256
#define MT   2            // m-tiles per block (register blocking over M)

// ---------------------------- utility kernels ------------------------------

__global__ void zero_kernel(float* __restrict__ buf, long n) {
    long i = (long)blockIdx.x * blockDim.x + threadIdx.x;
    if (i < n) buf[i] = 0.0f;
}

__global__ void add_kernel(const float* __restrict__ a, const float* __restrict__ b,
                           float* __restrict__ o, int n) {
    int i = blockIdx.x * blockDim.x + threadIdx.x;
    if (i < n) o[i] = a[i] + b[i];
}

__global__ void count_kernel(const int* __restrict__ dst, float* __restrict__ cnt, int e) {
    int i = blockIdx.x * blockDim.x + threadIdx.x;
    if (i < e) atomicAdd(&cnt[dst[i]], 1.0f);
}

// edge-parallel gather + scatter-add: one wave-chunk of threads per edge,
// float4 per thread (coalesced 16B loads, contiguous f32 atomics into L2)
template <int D>
__global__ void scatter_sum_kernel(const float* __restrict__ x,
                                   const int* __restrict__ src,
                                   const int* __restrict__ dst,
                                   float* __restrict__ sum) {
    const int CH = D / 4;
    long idx = (long)blockIdx.x * blockDim.x + threadIdx.x;
    long e = idx / CH;
    int  c = (int)(idx % CH);
    if (e >= EE) return;
    int s = src[e];
    int d = dst[e];
    float4 v = ((const float4*)(x + (long)s * D))[c];
    float* o = sum + (long)d * D + c * 4;
    atomicAdd(o + 0, v.x);
    atomicAdd(o + 1, v.y);
    atomicAdd(o + 2, v.z);
    atomicAdd(o + 3, v.w);
}

// sum -> mean (in place); isolated nodes divide by 1
template <int D>
__global__ void scale_kernel(float* __restrict__ buf, const float* __restrict__ cnt) {
    long i = (long)blockIdx.x * blockDim.x + threadIdx.x;
    if (i >= (long)NN * D) return;
    int row = (int)(i / D);
    buf[i] = buf[i] / fmaxf(cnt[row], 1.0f);
}

// --------------------------- layer-0 WMMA GEMM -----------------------------
// out[N,256] = relu( A0@B0.T + A1@B1.T + A2@B2.T + bias ), K=128 each panel.
// One 512-thread block (16 waves) per MT=2 m-tiles (32 rows); wave w owns
// n-tile w for both m-tiles. The three 32x128 A panels (48 KB) are staged in
// LDS once per block via GLOBAL_LOAD_ASYNC_TO_LDS_B128 (ASYNCcnt). Per
// k-step one global b64 B fragment feeds TWO WMMAs into independent
// accumulators -> no WMMA->WMMA RAW chain and half the B traffic.
// f32 16x16x4 fragment layout:
//   A: lane(0..15)->M=lane, VGPRv->K=v   | lane(16..31)->M=lane-16, K=2+v
//   B: lane(0..15)->N=lane, VGPRv->K=v   | lane(16..31)->N=lane-16, K=2+v
//   C: VGPRr: lanes0-15 -> (M=r, N=lane) | lanes16-31 -> (M=8+r, N=lane-16)
__global__ void __launch_bounds__(512)
gemm_l0_kernel(const float* __restrict__ A0, const float* __restrict__ B0,
               const float* __restrict__ A1, const float* __restrict__ B1,
               const float* __restrict__ A2, const float* __restrict__ B2,
               const float* __restrict__ bias, float* __restrict__ out) {
    __shared__ float At[3 * MT * 16 * DIN];   // 48 KB: 3 panels x 32 rows

    const int tid  = threadIdx.x;
    const int lane = tid & 31;
    const int wave = tid >> 5;                // 0..15 -> n-tile
    const int hi   = lane >> 4;               // 0: K-pair {0,1}, 1: K-pair {2,3}
    const int ml   = lane & 15;
    const int m0   = blockIdx.x * (16 * MT);
    const int n0   = wave * 16;

    const float* As[3] = {A0, A1, A2};
    const float* Bs[3] = {B0, B1, B2};

    // ---- stage A panels into LDS: one float4 per thread per (panel, tile) --
    // tile is 16x128 f32 = 512 float4; thread tid -> row tid>>5, float4 tid&31.
    // Tail block: clamp row to NN-1 (in-bounds read, value unused).
    {
        int r_in = tid >> 5;                  // 0..15 row inside tile
        int c4   = tid & 31;                  // float4 column 0..31
#pragma unroll
        for (int p = 0; p < 3; ++p) {
#pragma unroll
            for (int t = 0; t < MT; ++t) {
                int row = m0 + t * 16 + r_in;
                if (row >= NN) row = NN - 1;
                const float4* g = (const float4*)(As[p] + (long)row * DIN) + c4;
                float4*       l = ((float4*)At) + (p * MT + t) * 512 + tid;
#if HAVE_ASYNC_LDS
                __builtin_amdgcn_global_load_async_to_lds_b128(
                    (AS1 v4i*)g, (AS3 v4i*)l, 0, 0);
#else
                *l = *g;
#endif
            }
        }
#if HAVE_ASYNC_LDS
#if __has_builtin(__builtin_amdgcn_s_wait_asynccnt)
        __builtin_amdgcn_s_wait_asynccnt(0);
#else
        asm volatile("s_wait_asynccnt 0" ::: "memory");
#endif
#endif
        __syncthreads();
    }

    v8f c[MT] = {};
#pragma unroll
    for (int p = 0; p < 3; ++p) {
        const float* a_base = At + p * (MT * 16 * DIN) + ml * DIN + 2 * hi;  // LDS
        const float* b_row  = Bs[p] + (long)(n0 + ml) * DIN + 2 * hi;        // W row n
#pragma unroll
        for (int k = 0; k < DIN; k += 4) {
            v2f b = *(const v2f*)(b_row + k);
#pragma unroll
            for (int t = 0; t < MT; ++t) {
                v2f a = *(const v2f*)(a_base + t * (16 * DIN) + k);
                c[t] = __builtin_amdgcn_wmma_f32_16x16x4_f32(
                           false, a, false, b, (short)0, c[t], false, false);
            }
        }
    }

    const float bv = bias[n0 + ml];
#pragma unroll
    for (int t = 0; t < MT; ++t) {
#pragma unroll
        for (int r = 0; r < 8; ++r) {
            int row = m0 + t * 16 + r + 8 * hi;
            if (row < NN)
                out[(long)row * DH + n0 + ml] = fmaxf(c[t][r] + bv, 0.0f);
        }
    }
}

// --------------------------- layer-1 (dout = 2) ----------------------------
// o[m,0..1] = sum_p A_p[m,:] . B_p[0..1,:] + bias ; K=256. Thread per row.
__global__ void out_l1_kernel(const float* __restrict__ A0, const float* __restrict__ B0,
                              const float* __restrict__ A1, const float* __restrict__ B1,
                              const float* __restrict__ A2, const float* __restrict__ B2,
                              const float* __restrict__ bias, float* __restrict__ out) {
    int m = blockIdx.x * blockDim.x + threadIdx.x;
    if (m >= NN) return;
    float o0 = bias[0], o1 = bias[1];
    const float* As[3] = {A0, A1, A2};
    const float* Bs[3] = {B0, B1, B2};
#pragma unroll
    for (int p = 0; p < 3; ++p) {
        const float4* a  = (const float4*)(As[p] + (long)m * DH);
        const float4* w0 = (const float4*)(Bs[p]);
        const float4* w1 = (const float4*)(Bs[p] + DH);
#pragma unroll 8
        for (int k = 0; k < DH / 4; ++k) {
            float4 av = a[k], u = w0[k], v = w1[k];
            o0 += av.x * u.x + av.y * u.y + av.z * u.z + av.w * u.w;
            o1 += av.x * v.x + av.y * v.y + av.z * v.z + av.w * v.w;
        }
    }
    out[(long)m * 2 + 0] = o0;
    out[(long)m * 2 + 1] = o1;
}

// ------------------------------- launcher ----------------------------------

extern "C" void kernel_launch(void* const* d_in, const int* in_sizes, int n_in,
                              void* d_out, int out_size, void* d_ws, size_t ws_size,
                              hipStream_t stream) {
    (void)in_sizes; (void)n_in; (void)out_size; (void)ws_size;

    const float* x_tx = (const float*)d_in[0];
    const float* x_ad = (const float*)d_in[1];
    // relation order: tt, aa, at, ta
    const int* e[4] = {(const int*)d_in[2], (const int*)d_in[3],
                       (const int*)d_in[4], (const int*)d_in[5]};
    // layer 0 weights: Wl0_r, bl0_r, Wr0_r (r = tt, aa, at, ta)
    const float* Wl0[4] = {(const float*)d_in[6],  (const float*)d_in[9],
                           (const float*)d_in[12], (const float*)d_in[15]};
    const float* bl0[4] = {(const float*)d_in[7],  (const float*)d_in[10],
                           (const float*)d_in[13], (const float*)d_in[16]};
    const float* Wr0[4] = {(const float*)d_in[8],  (const float*)d_in[11],
                           (const float*)d_in[14], (const float*)d_in[17]};
    const float* Wl1[4] = {(const float*)d_in[18], (const float*)d_in[21],
                           (const float*)d_in[24], (const float*)d_in[27]};
    const float* bl1[4] = {(const float*)d_in[19], (const float*)d_in[22],
                           (const float*)d_in[25], (const float*)d_in[28]};
    const float* Wr1[4] = {(const float*)d_in[20], (const float*)d_in[23],
                           (const float*)d_in[26], (const float*)d_in[29]};

    // ---------------- workspace layout (floats) ----------------
    float* ws = (float*)d_ws;
    size_t off = 0;
    float* cnt[4];
    for (int r = 0; r < 4; ++r) { cnt[r] = ws + off; off += NN; }
    float* sum0[4];
    for (int r = 0; r < 4; ++r) { sum0[r] = ws + off; off += (size_t)NN * DIN; }
    float* h_tx = ws + off; off += (size_t)NN * DH;
    float* h_ad = ws + off; off += (size_t)NN * DH;
    float* sum1[4];
    for (int r = 0; r < 4; ++r) { sum1[r] = ws + off; off += (size_t)NN * DH; }
    float* wr0tx = ws + off; off += DH * DIN;
    float* wr0ad = ws + off; off += DH * DIN;
    float* b0tx  = ws + off; off += DH;
    float* b0ad  = ws + off; off += DH;
    float* wr1tx = ws + off; off += 2 * DH;
    float* wr1ad = ws + off; off += 2 * DH;
    float* b1tx  = ws + off; off += 2;
    float* b1ad  = ws + off; off += 2;

    const int T = 256;
    auto blocks = [](long n, int t) { return (int)((n + t - 1) / t); };

    // zero counts + layer-0 sums (one contiguous region)
    long z0 = 4L * NN + 4L * NN * DIN;
    zero_kernel<<<blocks(z0, T), T, 0, stream>>>(ws, z0);

    // per-relation in-degree counts (edge row 1 = destinations)
    for (int r = 0; r < 4; ++r)
        count_kernel<<<blocks(EE, T), T, 0, stream>>>(e[r] + EE, cnt[r], EE);

    // combined Wr / bias per destination type (tx: tt+at, addr: aa+ta)
    add_kernel<<<blocks(DH * DIN, T), T, 0, stream>>>(Wr0[0], Wr0[2], wr0tx, DH * DIN);
    add_kernel<<<blocks(DH * DIN, T), T, 0, stream>>>(Wr0[1], Wr0[3], wr0ad, DH * DIN);
    add_kernel<<<1, DH, 0, stream>>>(bl0[0], bl0[2], b0tx, DH);
    add_kernel<<<1, DH, 0, stream>>>(bl0[1], bl0[3], b0ad, DH);
    add_kernel<<<blocks(2 * DH, T), T, 0, stream>>>(Wr1[0], Wr1[2], wr1tx, 2 * DH);
    add_kernel<<<blocks(2 * DH, T), T, 0, stream>>>(Wr1[1], Wr1[3], wr1ad, 2 * DH);
    add_kernel<<<1, 2, 0, stream>>>(bl1[0], bl1[2], b1tx, 2);
    add_kernel<<<1, 2, 0, stream>>>(bl1[1], bl1[3], b1ad, 2);

    // ---------------- layer 0 aggregation ----------------
    const float* src0[4] = {x_tx, x_ad, x_ad, x_tx};  // tt, aa, at, ta sources
    long th0 = (long)EE * (DIN / 4);
    for (int r = 0; r < 4; ++r)
        scatter_sum_kernel<DIN><<<blocks(th0, T), T, 0, stream>>>(
            src0[r], e[r], e[r] + EE, sum0[r]);
    for (int r = 0; r < 4; ++r)
        scale_kernel<DIN><<<blocks((long)NN * DIN, T), T, 0, stream>>>(sum0[r], cnt[r]);

    // ---------------- layer 0 fused WMMA GEMM + bias + ReLU ----------------
    int g_gemm = (NN + 16 * MT - 1) / (16 * MT);
    gemm_l0_kernel<<<g_gemm, 512, 0, stream>>>(sum0[0], Wl0[0], sum0[2], Wl0[2],
                                               x_tx, wr0tx, b0tx, h_tx);
    gemm_l0_kernel<<<g_gemm, 512, 0, stream>>>(sum0[1], Wl0[1], sum0[3], Wl0[3],
                                               x_ad, wr0ad, b0ad, h_ad);

    // ---------------- layer 1 aggregation ----------------
    long z1 = 4L * NN * DH;
    zero_kernel<<<blocks(z1, T), T, 0, stream>>>(sum1[0], z1);
    const float* src1[4] = {h_tx, h_ad, h_ad, h_tx};
    long th1 = (long)EE * (DH / 4);
    for (int r = 0; r < 4; ++r)
        scatter_sum_kernel<DH><<<blocks(th1, T), T, 0, stream>>>(
            src1[r], e[r], e[r] + EE, sum1[r]);
    for (int r = 0; r < 4; ++r)
        scale_kernel<DH><<<blocks((long)NN * DH, T), T, 0, stream>>>(sum1[r], cnt[r]);

    // ---------------- layer 1 output projection (dout = 2) ----------------
    float* o_tx = (float*)d_out;
    float* o_ad = (float*)d_out + 2L * NN;
    out_l1_kernel<<<blocks(NN, T), T, 0, stream>>>(sum1[0], Wl1[0], sum1[2], Wl1[2],
                                                   h_tx, wr1tx, b1tx, o_tx);
    out_l1_kernel<<<blocks(NN, T), T, 0, stream>>>(sum1[1], Wl1[1], sum1[3], Wl1[3],
                                                   h_ad, wr1ad, b1ad, o_ad);
}